// GuidedAttentionLoss_970662609263
// MI455X (gfx1250) — compile-verified
//
#include <hip/hip_runtime.h>

#define T_DEC 2048
#define T_ENC 512
#define NBATCH 64
#define ROWS (NBATCH * T_DEC)
#define G_STEPS 20000
#define DEPTH 4            // async pipeline depth (row buffers)

typedef __attribute__((ext_vector_type(2))) float v2f;
typedef __attribute__((ext_vector_type(8))) float v8f;

// ln(0.99995)
#define LN_GAMMA (-5.000125002083345e-05f)

__device__ __forceinline__ int next_valid_row(int r, int stride, const int* __restrict__ Fp) {
    // skip rows whose decoder index f >= target_length[b]; block-uniform
    while (r < ROWS) {
        if ((r & (T_DEC - 1)) < Fp[r >> 11]) break;
        r += stride;
    }
    return r;
}

__global__ __launch_bounds__(128) void ga_main(const float* __restrict__ A,
                                               const int* __restrict__ Lp,   // input_lengths  [B]
                                               const int* __restrict__ Fp,   // target_lengths [B]
                                               const int* __restrict__ stepp,
                                               float* __restrict__ ws) {
    __shared__ float buf[DEPTH][T_ENC];   // 4 x 2KB row tiles
    __shared__ float wsum[4];

    const int tid = threadIdx.x;
    const int l0  = tid * 4;              // 128 lanes x float4 = one 512-float row
    float acc = 0.0f;

    const int step = stepp[0];
    if (step <= G_STEPS) {
        const float g = __expf(LN_GAMMA * (float)step);
        const float neg_inv_denom = -1.0f / (2.0f * g * g);
        const int stride = gridDim.x;

        // Async issue with per-lane EXEC masking (lanes with l0 >= L load nothing).
        // EXEC is swapped inside the asm so the instruction is *always* issued at
        // the wave level -> per-wave ASYNCcnt matches the uniform wait immediates.
        // (HW elides a VMEM op only when EXEC==0 && ASYNCcnt==0, i.e. when all
        // prior loads already retired -- still safe.)
        auto issue = [&](int rrow, int bi) {
            const int b  = rrow >> 11;
            const int Li = Lp[b];
            unsigned mask = (unsigned)__ballot(l0 < Li);
            unsigned long long gp = (unsigned long long)(A + ((size_t)rrow * T_ENC + l0));
            unsigned lds = (unsigned)(size_t)&buf[bi][l0];
            unsigned saved;
            asm volatile("s_wait_dscnt 0x0\n"            // prior ds reads of this buf retired
                         "s_mov_b32 %0, exec_lo\n"
                         "s_mov_b32 exec_lo, %3\n"
                         "global_load_async_to_lds_b128 %1, %2, off\n"
                         "s_mov_b32 exec_lo, %0"
                         : "=&s"(saved)
                         : "v"(lds), "v"(gp), "s"(mask)
                         : "memory");
        };

        auto compute = [&](int rrow, int bi) {
            const int b = rrow >> 11;
            const int f = rrow & (T_DEC - 1);
            const float Ff   = (float)Fp[b];
            const float Lf   = (float)Lp[b];
            const float invF = 1.0f / Ff;
            const float invL = 1.0f / Lf;
            const float fF   = (float)f * invF;
            const float4 av = *(const float4*)&buf[bi][l0];
            float rsum = 0.0f;
            #pragma unroll
            for (int j = 0; j < 4; ++j) {
                const float lv = (float)(l0 + j);
                const float x  = lv * invL - fF;
                const float w  = 1.0f - __expf(x * x * neg_inv_denom);
                rsum += (lv < Lf) ? w * (&av.x)[j] : 0.0f;   // masks stale lanes too
            }
            acc += rsum * invF;   // fold per-sample 1/F here
        };

        int rn[DEPTH];
        int r = next_valid_row(blockIdx.x, stride, Fp);
        #pragma unroll
        for (int d = 0; d < DEPTH; ++d) {     // fill pipeline
            rn[d] = r;
            if (r < ROWS) {
                issue(r, d);
                r = next_valid_row(r + stride, stride, Fp);
            }
        }
        for (;;) {
            #pragma unroll
            for (int d = 0; d < DEPTH; ++d) {
                if (rn[d] >= ROWS) goto done;
                if (r < ROWS) {
                    // steady state: queue is exactly DEPTH deep -> oldest done at <=3
                    asm volatile("s_wait_asynccnt 0x3" ::: "memory");
                    compute(rn[d], d);
                    issue(r, d);
                    rn[d] = r;
                    r = next_valid_row(r + stride, stride, Fp);
                } else {
                    // drain: force everything outstanding to complete
                    asm volatile("s_wait_asynccnt 0x0" ::: "memory");
                    compute(rn[d], d);
                    rn[d] = ROWS;
                }
            }
        }
        done: ;
    }

    // ---- wave32 reduction via V_WMMA_F32_16X16X4_F32 against all-ones B ----
    // A(16x4): lane m (0-15) -> A[m][0]=acc, lane m+16 -> A[m][2]=acc, K=1,3 zero.
    // D[m][n] = rowsum(m); summing the 8 D VGPRs per lane gives half-M partials,
    // one xor-16 shuffle completes the 32-lane sum. (EXEC all-1s here: no divergence.)
    {
        v2f a2; a2.x = acc;  a2.y = 0.0f;
        v2f b2; b2.x = 1.0f; b2.y = 1.0f;
        v8f c = {};
        c = __builtin_amdgcn_wmma_f32_16x16x4_f32(
                /*neg_a=*/false, a2, /*neg_b=*/false, b2,
                /*c_mod=*/(short)0, c, /*reuse_a=*/false, /*reuse_b=*/false);
        float wtot = ((c[0] + c[1]) + (c[2] + c[3])) + ((c[4] + c[5]) + (c[6] + c[7]));
        wtot += __shfl_xor(wtot, 16, 32);

        if ((tid & 31) == 0) wsum[tid >> 5] = wtot;
        __syncthreads();
        if (tid == 0) ws[blockIdx.x] = (wsum[0] + wsum[1]) + (wsum[2] + wsum[3]);
    }
}

__global__ __launch_bounds__(32) void ga_final(const float* __restrict__ ws,
                                               const int* __restrict__ stepp,
                                               float* __restrict__ out, int n) {
    const int step = stepp[0];
    if (step > G_STEPS) {
        if (threadIdx.x == 0) out[0] = 0.0f;
        return;
    }
    float s = 0.0f;
    for (int i = threadIdx.x; i < n; i += 32) s += ws[i];   // fixed-order -> deterministic
    #pragma unroll
    for (int off = 16; off > 0; off >>= 1) s += __shfl_down(s, off, 32);
    if (threadIdx.x == 0) out[0] = s * (1.0f / (float)NBATCH);
}

extern "C" void kernel_launch(void* const* d_in, const int* in_sizes, int n_in,
                              void* d_out, int out_size, void* d_ws, size_t ws_size,
                              hipStream_t stream) {
    const float* A     = (const float*)d_in[0];   // alignments [64,2048,512] f32
    const int*   Lp    = (const int*)d_in[1];     // input_lengths  [64]
    const int*   Fp    = (const int*)d_in[2];     // target_lengths [64]
    const int*   stepp = (const int*)d_in[3];     // global_step (scalar)
    float* out = (float*)d_out;
    float* ws  = (float*)d_ws;

    int nblk = 8192;
    size_t cap = ws_size / sizeof(float);
    if ((size_t)nblk > cap) nblk = (int)cap;
    if (nblk < 1) nblk = 1;

    ga_main <<<nblk, 128, 0, stream>>>(A, Lp, Fp, stepp, ws);
    ga_final<<<1,    32,  0, stream>>>(ws, stepp, out, nblk);
}